// LP1_89816356094074
// MI455X (gfx1250) — compile-verified
//
#include <hip/hip_runtime.h>
#include <hip/hip_bf16.h>
#include <math.h>

typedef __attribute__((ext_vector_type(2))) float v2f;
typedef __attribute__((ext_vector_type(8))) float v8f;

#define EPS 1e-4f
#define RMAX 32
#define WPB 4   // waves per block for WMMA kernels (wave32)

#define WMMA_F32(acc, a, b) \
    __builtin_amdgcn_wmma_f32_16x16x4_f32(false, (a), false, (b), (short)0, (acc), false, false)

// ---------------------------------------------------------------------------
// Residual + reweight:  w2 = mask * rsqrt((U@V - X)^2 + eps)
// U:[m,r], V:[r,n], X/w2:[m,n]. One wave per 16x16 tile; 4 waves/block.
// Interior tiles: unconditional loads. Edge tiles: clamped index + v_cndmask
// zeroing (no EXEC masking -> EXEC stays all-1s for WMMA).
// ---------------------------------------------------------------------------
__global__ __launch_bounds__(32 * WPB)
void resid_w2_wmma(const float* __restrict__ X, const float* __restrict__ U,
                   const float* __restrict__ V, float* __restrict__ w2,
                   int m, int n, int r, const int* __restrict__ layer,
                   int use_layer)
{
    const int lane = threadIdx.x & 31;
    const long tile = (long)blockIdx.x * WPB + (threadIdx.x >> 5);
    const int ntile = (n + 15) >> 4;
    const long ttot = (long)((m + 15) >> 4) * ntile;
    if (tile >= ttot) return;                    // wave-uniform
    const int ti = (int)(tile / ntile), tj = (int)(tile % ntile);
    const int i0 = ti << 4, j0 = tj << 4;
    const int l15 = lane & 15;
    const int hi  = lane >> 4;                   // 0: lanes 0-15, 1: lanes 16-31
    const int arow = i0 + l15;
    const int bcol = j0 + l15;
    const int r4 = r & ~3;

    v8f acc = {};
    if ((i0 + 16 <= m) && (j0 + 16 <= n)) {      // wave-uniform fast path
        const float* __restrict__ urow = U + (long)arow * r;
        int kk = 0;
        for (; kk < r4; kk += 4) {
            const int ak = kk + (hi << 1);
            v2f a, b;
            a.x = urow[ak];
            a.y = urow[ak + 1];
            b.x = V[(long)ak * n + bcol];
            b.y = V[(long)(ak + 1) * n + bcol];
            acc = WMMA_F32(acc, a, b);
        }
        if (kk < r) {                            // K remainder: clamp + select
            const int ak = kk + (hi << 1);
            const int k0 = (ak     < r) ? ak     : r - 1;
            const int k1 = (ak + 1 < r) ? ak + 1 : r - 1;
            const float u0 = urow[k0], u1 = urow[k1];
            const float p0 = V[(long)k0 * n + bcol], p1 = V[(long)k1 * n + bcol];
            v2f a, b;
            a.x = (ak     < r) ? u0 : 0.f;
            a.y = (ak + 1 < r) ? u1 : 0.f;
            b.x = (ak     < r) ? p0 : 0.f;
            b.y = (ak + 1 < r) ? p1 : 0.f;
            acc = WMMA_F32(acc, a, b);
        }
    } else {                                     // edge tile: clamp + select
        const int ar_c = (arow < m) ? arow : m - 1;
        const int bc_c = (bcol < n) ? bcol : n - 1;
        const bool av = arow < m, bv = bcol < n;
        for (int kk = 0; kk < r; kk += 4) {
            const int ak = kk + (hi << 1);
            const int k0 = (ak     < r) ? ak     : r - 1;
            const int k1 = (ak + 1 < r) ? ak + 1 : r - 1;
            const float u0 = U[(long)ar_c * r + k0];
            const float u1 = U[(long)ar_c * r + k1];
            const float p0 = V[(long)k0 * n + bc_c];
            const float p1 = V[(long)k1 * n + bc_c];
            v2f a, b;
            a.x = (av && ak     < r) ? u0 : 0.f;
            a.y = (av && ak + 1 < r) ? u1 : 0.f;
            b.x = (bv && ak     < r) ? p0 : 0.f;
            b.y = (bv && ak + 1 < r) ? p1 : 0.f;
            acc = WMMA_F32(acc, a, b);
        }
    }

    const bool lay1 = (use_layer != 0) && (layer[0] == 1);
    for (int v = 0; v < 8; ++v) {
        const int row = i0 + v + (hi << 3);
        const int col = j0 + l15;
        if (row < m && col < n) {
            const float x = X[(long)row * n + col];
            const float mask = (x != 0.f) ? 1.f : 0.f;
            float out;
            if (lay1) {
                out = mask;
            } else {
                const float ks = acc[v] - x;
                out = mask * rsqrtf(ks * ks + EPS);
            }
            w2[(long)row * n + col] = out;
        }
    }
}

// ---------------------------------------------------------------------------
// Batched weighted Gram:  A[b,r,s] = sum_t F(r,t)*w2(b,t)*F(s,t)
// F(rr,t) = F[rr*frs + t*fts];  w2(b,t) = W2[b*wbs + t*wts]
// One wave per batch element; 4 waves/block; clamp+select (no EXEC masking).
// ---------------------------------------------------------------------------
__global__ __launch_bounds__(32 * WPB)
void gram_wmma(const float* __restrict__ F, const float* __restrict__ W2,
               float* __restrict__ A, int T, int r, int nbatch,
               long frs, long fts, long wbs, long wts)
{
    const int lane = threadIdx.x & 31;
    const long b = (long)blockIdx.x * WPB + (threadIdx.x >> 5);
    if (b >= nbatch) return;                     // wave-uniform
    const int l15 = lane & 15;
    const int hi  = lane >> 4;
    const long wbase = b * wbs;
    const long rr2 = (long)r * r;
    const int T4 = T & ~3;

    for (int rb = 0; rb < r; rb += 16)
    for (int sb = 0; sb < r; sb += 16) {
        const int arow = rb + l15;
        const int bcol = sb + l15;
        const int ar_c = (arow < r) ? arow : 0;
        const int bc_c = (bcol < r) ? bcol : 0;
        const bool av = arow < r, bv = bcol < r;
        v8f acc = {};
        int tt = 0;
        for (; tt < T4; tt += 4) {
            const int ak = tt + (hi << 1);
            const float w0 = W2[wbase + (long)ak * wts];
            const float w1 = W2[wbase + (long)(ak + 1) * wts];
            const float f0 = F[(long)ar_c * frs + (long)ak * fts];
            const float f1 = F[(long)ar_c * frs + (long)(ak + 1) * fts];
            const float g0 = F[(long)bc_c * frs + (long)ak * fts];
            const float g1 = F[(long)bc_c * frs + (long)(ak + 1) * fts];
            v2f a, bb;
            a.x  = av ? f0 * w0 : 0.f;
            a.y  = av ? f1 * w1 : 0.f;
            bb.x = bv ? g0 : 0.f;
            bb.y = bv ? g1 : 0.f;
            acc = WMMA_F32(acc, a, bb);
        }
        if (tt < T) {                            // T remainder
            const int ak = tt + (hi << 1);
            const int k0 = (ak     < T) ? ak     : T - 1;
            const int k1 = (ak + 1 < T) ? ak + 1 : T - 1;
            const float w0 = W2[wbase + (long)k0 * wts];
            const float w1 = W2[wbase + (long)k1 * wts];
            const float f0 = F[(long)ar_c * frs + (long)k0 * fts];
            const float f1 = F[(long)ar_c * frs + (long)k1 * fts];
            const float g0 = F[(long)bc_c * frs + (long)k0 * fts];
            const float g1 = F[(long)bc_c * frs + (long)k1 * fts];
            v2f a, bb;
            a.x  = (av && ak     < T) ? f0 * w0 : 0.f;
            a.y  = (av && ak + 1 < T) ? f1 * w1 : 0.f;
            bb.x = (bv && ak     < T) ? g0 : 0.f;
            bb.y = (bv && ak + 1 < T) ? g1 : 0.f;
            acc = WMMA_F32(acc, a, bb);
        }
        for (int v = 0; v < 8; ++v) {
            const int row = rb + v + (hi << 3);
            const int col = sb + l15;
            if (row < r && col < r)
                A[b * rr2 + (long)row * r + col] = acc[v];
        }
    }
}

// ---------------------------------------------------------------------------
// rhs[b,rr] = sum_t F(rr,t) * w2(b,t) * X(b,t)  as WMMA GEMM:
// D[rank, batch16] = F[rank,T] @ (w2*X)[T, batch16]. One wave per 16-batch
// tile; 4 waves/block. rhs stored batch-major: rhs[b*r + rr].
// ---------------------------------------------------------------------------
__global__ __launch_bounds__(32 * WPB)
void rhs_wmma(const float* __restrict__ F, const float* __restrict__ W2,
              const float* __restrict__ X, float* __restrict__ rhs,
              int T, int r, int nbatch,
              long frs, long fts, long wbs, long wts)
{
    const int lane = threadIdx.x & 31;
    const long tile = (long)blockIdx.x * WPB + (threadIdx.x >> 5);
    const long ntiles = ((long)nbatch + 15) >> 4;
    if (tile >= ntiles) return;                  // wave-uniform
    const int b0 = (int)(tile << 4);
    const int l15 = lane & 15;
    const int hi  = lane >> 4;
    const int bcol = b0 + l15;
    const int bc_c = (bcol < nbatch) ? bcol : 0;
    const bool bv = bcol < nbatch;
    const int T4 = T & ~3;

    for (int rb = 0; rb < r; rb += 16) {
        const int arow = rb + l15;
        const int ar_c = (arow < r) ? arow : 0;
        const bool av = arow < r;
        v8f acc = {};
        int tt = 0;
        for (; tt < T4; tt += 4) {
            const int ak = tt + (hi << 1);
            const float f0 = F[(long)ar_c * frs + (long)ak * fts];
            const float f1 = F[(long)ar_c * frs + (long)(ak + 1) * fts];
            const long o0 = (long)bc_c * wbs + (long)ak * wts;
            const long o1 = (long)bc_c * wbs + (long)(ak + 1) * wts;
            const float g0 = W2[o0] * X[o0];
            const float g1 = W2[o1] * X[o1];
            v2f a, bb;
            a.x  = av ? f0 : 0.f;
            a.y  = av ? f1 : 0.f;
            bb.x = bv ? g0 : 0.f;
            bb.y = bv ? g1 : 0.f;
            acc = WMMA_F32(acc, a, bb);
        }
        if (tt < T) {
            const int ak = tt + (hi << 1);
            const int k0 = (ak     < T) ? ak     : T - 1;
            const int k1 = (ak + 1 < T) ? ak + 1 : T - 1;
            const float f0 = F[(long)ar_c * frs + (long)k0 * fts];
            const float f1 = F[(long)ar_c * frs + (long)k1 * fts];
            const long o0 = (long)bc_c * wbs + (long)k0 * wts;
            const long o1 = (long)bc_c * wbs + (long)k1 * wts;
            const float g0 = W2[o0] * X[o0];
            const float g1 = W2[o1] * X[o1];
            v2f a, bb;
            a.x  = (av && ak     < T) ? f0 : 0.f;
            a.y  = (av && ak + 1 < T) ? f1 : 0.f;
            bb.x = (bv && ak     < T) ? g0 : 0.f;
            bb.y = (bv && ak + 1 < T) ? g1 : 0.f;
            acc = WMMA_F32(acc, a, bb);
        }
        for (int v = 0; v < 8; ++v) {
            const int row = rb + v + (hi << 3);
            const int col = b0 + l15;
            if (row < r && col < nbatch)
                rhs[(long)col * r + row] = acc[v];
        }
    }
}

// ---------------------------------------------------------------------------
// Damped Gauss-Jordan inverse of each r x r Gram matrix (approximates pinv;
// near-zero pivots skipped). One thread per batch element.
// ---------------------------------------------------------------------------
__global__ void pinv_kernel(const float* __restrict__ A, float* __restrict__ Ainv,
                            int r, int nbatch)
{
    const long b = (long)blockIdx.x * blockDim.x + threadIdx.x;
    if (b >= nbatch) return;
    const int rr = (r > RMAX) ? RMAX : r;
    const long rr2 = (long)r * r;
    float M[RMAX][RMAX];
    float I[RMAX][RMAX];

    float tr = 0.f;
    for (int i = 0; i < rr; ++i) tr += A[b * rr2 + (long)i * r + i];
    const float lam = 1e-7f * (tr / (float)rr) + 1e-20f;

    for (int i = 0; i < rr; ++i)
        for (int j = 0; j < rr; ++j) {
            M[i][j] = A[b * rr2 + (long)i * r + j] + ((i == j) ? lam : 0.f);
            I[i][j] = (i == j) ? 1.f : 0.f;
        }

    for (int kcol = 0; kcol < rr; ++kcol) {
        int piv = kcol;
        float best = fabsf(M[kcol][kcol]);
        for (int i = kcol + 1; i < rr; ++i) {
            const float v = fabsf(M[i][kcol]);
            if (v > best) { best = v; piv = i; }
        }
        if (best < 1e-30f) continue;
        if (piv != kcol)
            for (int j = 0; j < rr; ++j) {
                float t = M[kcol][j]; M[kcol][j] = M[piv][j]; M[piv][j] = t;
                t = I[kcol][j]; I[kcol][j] = I[piv][j]; I[piv][j] = t;
            }
        const float inv = 1.f / M[kcol][kcol];
        for (int j = 0; j < rr; ++j) { M[kcol][j] *= inv; I[kcol][j] *= inv; }
        for (int i = 0; i < rr; ++i) {
            if (i == kcol) continue;
            const float f = M[i][kcol];
            if (f == 0.f) continue;
            for (int j = 0; j < rr; ++j) {
                M[i][j] -= f * M[kcol][j];
                I[i][j] -= f * I[kcol][j];
            }
        }
    }

    for (int i = 0; i < r; ++i)
        for (int j = 0; j < r; ++j)
            Ainv[b * rr2 + (long)i * r + j] = (i < rr && j < rr) ? I[i][j] : 0.f;
}

// ---------------------------------------------------------------------------
// Per-sample SAME 2D cross-correlation (lax.conv: no flip, zero pad = k/2).
// ---------------------------------------------------------------------------
__global__ void conv_kernel(const float* __restrict__ Xin, const float* __restrict__ K,
                            float* __restrict__ Xout, int r, int k, int nbatch)
{
    const long idx = (long)blockIdx.x * blockDim.x + threadIdx.x;
    const long rr2 = (long)r * r;
    if (idx >= (long)nbatch * rr2) return;
    const long b = idx / rr2;
    const int rem = (int)(idx % rr2);
    const int a = rem / r, c = rem % r;
    const int pad = k >> 1;
    const long kbase = b * (long)k * k;
    float s = 0.f;
    for (int p = 0; p < k; ++p) {
        const int ia = a + p - pad;
        if (ia < 0 || ia >= r) continue;
        for (int q = 0; q < k; ++q) {
            const int ic = c + q - pad;
            if (ic < 0 || ic >= r) continue;
            s += Xin[b * rr2 + (long)ia * r + ic] * K[kbase + (long)p * k + q];
        }
    }
    Xout[idx] = s;
}

// ---------------------------------------------------------------------------
// beta[b,rr] = sum_s XcSel(b,rr,s) * rhs[b,s]
//   transXc=0 : Xc[b, rr, s]  (cols 'jrs,js->jr');  =1 : Xc[b, s, rr] (rows)
// out[b*obs + rr*ors]: cols -> Vcur[rr*n+j] (obs=1,ors=n); rows -> Ubuf[i*r+rr]
// ---------------------------------------------------------------------------
__global__ void beta_kernel(const float* __restrict__ Xc, const float* __restrict__ rhs,
                            float* __restrict__ out, int r, int nbatch, int transXc,
                            long obs, long ors)
{
    const long idx = (long)blockIdx.x * blockDim.x + threadIdx.x;
    if (idx >= (long)nbatch * r) return;
    const long b = idx / r;
    const int rr = (int)(idx % r);
    const long rr2 = (long)r * r;
    float s = 0.f;
    for (int si = 0; si < r; ++si) {
        const float xv = transXc ? Xc[b * rr2 + (long)si * r + rr]
                                 : Xc[b * rr2 + (long)rr * r + si];
        s += xv * rhs[b * r + si];
    }
    out[b * obs + (long)rr * ors] = s;
}

__global__ void finalize_rows(const float* __restrict__ Ubuf, const float* __restrict__ Uorig,
                              const float* __restrict__ X, float* __restrict__ out,
                              int m, int n, int r)
{
    const long idx = (long)blockIdx.x * blockDim.x + threadIdx.x;
    if (idx >= (long)m * r) return;
    const long i = idx / r;
    bool has = false;
    for (int j = 0; j < n; ++j)
        if (X[i * (long)n + j] != 0.f) { has = true; break; }
    out[idx] = has ? Ubuf[idx] : Uorig[idx];
}

__global__ void copy_f32(const float* __restrict__ src, float* __restrict__ dst, long count)
{
    const long idx = (long)blockIdx.x * blockDim.x + threadIdx.x;
    if (idx < count) dst[idx] = src[idx];
}

// ---------------------------------------------------------------------------

static inline unsigned cdiv_u(long a, long b) { return (unsigned)((a + b - 1) / b); }

extern "C" void kernel_launch(void* const* d_in, const int* in_sizes, int n_in,
                              void* d_out, int out_size, void* d_ws, size_t ws_size,
                              hipStream_t stream)
{
    (void)n_in; (void)out_size; (void)ws_size;
    const float* X  = (const float*)d_in[0];
    const float* U  = (const float*)d_in[1];
    const float* V  = (const float*)d_in[2];
    const float* Kw = (const float*)d_in[3];
    const int* layer = (const int*)d_in[4];

    const long sx = in_sizes[0], su = in_sizes[1], sv = in_sizes[2], sk = in_sizes[3];
    long r = (long)(sqrt((double)su * (double)sv / (double)sx) + 0.5);
    if (r < 1) r = 1;
    const long m = su / r;
    const long n = sv / r;
    const long B = m + n - 1;
    long k = (long)(sqrt((double)sk / (double)(B > 0 ? B : 1)) + 0.5);
    if (k < 1) k = 1;
    const long maxb = (m > n) ? m : n;

    float* ws   = (float*)d_ws;
    float* w2   = ws;  ws += m * n;
    float* Vcur = ws;  ws += r * n;
    float* Ubuf = ws;  ws += m * r;
    float* A    = ws;  ws += maxb * r * r;
    float* Ainv = ws;  ws += maxb * r * r;
    float* Xc   = ws;  ws += maxb * r * r;
    float* rhs  = ws;  ws += maxb * r;

    const long mt = (m + 15) >> 4, nt = (n + 15) >> 4;
    const dim3 blk(32 * WPB);
    const unsigned gResid = cdiv_u(mt * nt, WPB);
    const unsigned gGramC = cdiv_u(n, WPB);
    const unsigned gGramR = cdiv_u(m, WPB);
    const unsigned gRhsC  = cdiv_u((n + 15) >> 4, WPB);
    const unsigned gRhsR  = cdiv_u((m + 15) >> 4, WPB);

    // ======================== columns phase (lp1v) ========================
    copy_f32<<<cdiv_u(r * n, 256), 256, 0, stream>>>(V, Vcur, r * n);
    resid_w2_wmma<<<gResid, blk, 0, stream>>>(X, U, Vcur, w2,
                                              (int)m, (int)n, (int)r, layer, 1);
    for (int it = 0; it < 3; ++it) {
        gram_wmma<<<gGramC, blk, 0, stream>>>(U, w2, A, (int)m, (int)r, (int)n,
                                              1L, r, 1L, n);
        pinv_kernel<<<cdiv_u(n, 64), 64, 0, stream>>>(A, Ainv, (int)r, (int)n);
        conv_kernel<<<cdiv_u(n * r * r, 256), 256, 0, stream>>>(Ainv, Kw, Xc,
                                                                (int)r, (int)k, (int)n);
        rhs_wmma<<<gRhsC, blk, 0, stream>>>(U, w2, X, rhs,
                                            (int)m, (int)r, (int)n, 1L, r, 1L, n);
        beta_kernel<<<cdiv_u(n * r, 256), 256, 0, stream>>>(Xc, rhs, Vcur,
                                                            (int)r, (int)n, 0, 1L, n);
        if (it < 2)
            resid_w2_wmma<<<gResid, blk, 0, stream>>>(X, U, Vcur, w2,
                                                      (int)m, (int)n, (int)r, layer, 0);
    }

    // ========================= rows phase (lp1u) ==========================
    copy_f32<<<cdiv_u(m * r, 256), 256, 0, stream>>>(U, Ubuf, m * r);
    resid_w2_wmma<<<gResid, blk, 0, stream>>>(X, Ubuf, Vcur, w2,
                                              (int)m, (int)n, (int)r, layer, 0);
    const float* Kr = Kw + (size_t)(n - 1) * k * k;
    for (int it = 0; it < 3; ++it) {
        gram_wmma<<<gGramR, blk, 0, stream>>>(Vcur, w2, A, (int)n, (int)r, (int)m,
                                              n, 1L, n, 1L);
        pinv_kernel<<<cdiv_u(m, 64), 64, 0, stream>>>(A, Ainv, (int)r, (int)m);
        conv_kernel<<<cdiv_u(m * r * r, 256), 256, 0, stream>>>(Ainv, Kr, Xc,
                                                                (int)r, (int)k, (int)m);
        rhs_wmma<<<gRhsR, blk, 0, stream>>>(Vcur, w2, X, rhs,
                                            (int)n, (int)r, (int)m, n, 1L, n, 1L);
        beta_kernel<<<cdiv_u(m * r, 256), 256, 0, stream>>>(Xc, rhs, Ubuf,
                                                            (int)r, (int)m, 1, r, 1L);
        if (it < 2)
            resid_w2_wmma<<<gResid, blk, 0, stream>>>(X, Ubuf, Vcur, w2,
                                                      (int)m, (int)n, (int)r, layer, 0);
    }

    // =========================== outputs ==================================
    float* outU = (float*)d_out;          // U_new [m, r]
    float* outV = outU + m * r;           // V_new [r, n]
    finalize_rows<<<cdiv_u(m * r, 256), 256, 0, stream>>>(Ubuf, U, X, outU,
                                                          (int)m, (int)n, (int)r);
    copy_f32<<<cdiv_u(r * n, 256), 256, 0, stream>>>(Vcur, outV, r * n);
}